// SetConv_32435593019489
// MI455X (gfx1250) — compile-verified
//
#include <hip/hip_runtime.h>
#include <hip/hip_bf16.h>

typedef __attribute__((ext_vector_type(2))) float v2f;
typedef __attribute__((ext_vector_type(8))) float v8f;

#define LOG2E 1.44269504088896340736f

// Problem constants from the reference
#define B  8
#define N  1024
#define M  1024
#define C  16
#define O  32
#define ROWS (B*M)          // 8192 flattened (b,m) rows

// -----------------------------------------------------------------------------
// Kernel 1: RBF-weighted accumulation over n, split into n-chunks.
// partial[s][b*M+m][c] = sum_{n in chunk s} exp2(k2[c] * (x[b,n]-t[b,m])^2) * y[b,n,c]
// where k2[c] = -0.5 * exp(-2*sigma[c]) * log2(e)   (so exp2(k2*d) == exp(-0.5*d/scale^2))
// One thread per m; (x,y) n-chunks staged in LDS (wave-uniform broadcast reads).
// -----------------------------------------------------------------------------
__global__ __launch_bounds__(256) void rbf_partial_kernel(
    const float* __restrict__ x,      // (B, N, 1)
    const float* __restrict__ y,      // (B, N, C)
    const float* __restrict__ t,      // (B, M, 1)
    const float* __restrict__ sigma,  // (C)
    float* __restrict__ partial,      // (nsplit, B, M, C)
    int chunk)                        // N / nsplit
{
    __shared__ __align__(16) float xs[128];
    __shared__ __align__(16) float ys[128 * C];

    const int tid = threadIdx.x;
    const int m   = blockIdx.x * 256 + tid;   // gridDim.x = M/256 = 4
    const int b   = blockIdx.y;               // B
    const int s   = blockIdx.z;               // nsplit

    // Per-channel exponent coefficients (scalar/broadcast loads; trivial cost).
    float k2[C];
#pragma unroll
    for (int c = 0; c < C; ++c) {
        float sg = sigma[c];
        // 1/scale^2 = exp(-2*sigma) computed via v_exp_f32
        float inv_s2 = __builtin_amdgcn_exp2f(-2.0f * sg * LOG2E);
        k2[c] = -0.5f * inv_s2 * LOG2E;
    }

    const float tm = t[b * M + m];

    float acc[C];
#pragma unroll
    for (int c = 0; c < C; ++c) acc[c] = 0.0f;

    const int n_begin = s * chunk;
    const int n_end   = n_begin + chunk;

    for (int n0 = n_begin; n0 < n_end; n0 += 128) {
        // Cooperative stage: 128 x-values + 128*16 y-values (coalesced b128 loads)
        if (tid < 128) xs[tid] = x[b * N + n0 + tid];
        {
            const float4* ysrc = (const float4*)(y + (size_t)(b * N + n0) * C);
            float4* ydst = (float4*)ys;
#pragma unroll
            for (int i = tid; i < 128 * C / 4; i += 256) ydst[i] = ysrc[i];
        }
        __syncthreads();

        for (int j = 0; j < 128; ++j) {
            float d = xs[j] - tm;
            d = d * d;
            // 16 v_exp_f32 + 16 FMA per (n,m) pair; ds reads are broadcast.
#pragma unroll
            for (int c = 0; c < C; ++c) {
                acc[c] += __builtin_amdgcn_exp2f(d * k2[c]) * ys[j * C + c];
            }
        }
        __syncthreads();
    }

    // Vectorized partial store: (s, b*M+m, 0..15)
    float4* pdst = (float4*)(partial + ((size_t)(s * B + b) * M + m) * C);
    pdst[0] = make_float4(acc[0],  acc[1],  acc[2],  acc[3]);
    pdst[1] = make_float4(acc[4],  acc[5],  acc[6],  acc[7]);
    pdst[2] = make_float4(acc[8],  acc[9],  acc[10], acc[11]);
    pdst[3] = make_float4(acc[12], acc[13], acc[14], acc[15]);
}

// -----------------------------------------------------------------------------
// Kernel 2: deterministic partial reduction + (8192x16)@(16x32) projection on
// V_WMMA_F32_16X16X4_F32 (full f32 — matches reference precision), + bias.
// Block = 256 threads (8 waves), 128 rows/block. Each wave: one 16-row tile,
// two 16-col o-tiles, K=16 => 4 chained WMMAs per o-tile.
// Operand packing per CDNA5 ISA 7.12.2:
//   A (16x4, MxK): M = lane%16, K = kk + comp + 2*(lane>=16)
//   B (4x16, KxN): N = lane%16, K = kk + comp + 2*(lane>=16)
//   D (16x16):     M = vgpr + 8*(lane>=16), N = lane%16
// -----------------------------------------------------------------------------
__global__ __launch_bounds__(256) void reduce_wmma_kernel(
    const float* __restrict__ partial, // (nsplit, ROWS, C)
    const float* __restrict__ W,       // (O, C) row-major
    const float* __restrict__ bias,    // (O)
    float* __restrict__ out,           // (ROWS, O)
    int nsplit)
{
    __shared__ __align__(16) float yt[128 * C];

    const int tid  = threadIdx.x;
    const int row0 = blockIdx.x * 128;

    // Deterministic reduction of n-split partials (fixed summation order).
    for (int i = tid; i < 128 * C; i += 256) {
        float ssum = 0.0f;
        for (int s2 = 0; s2 < nsplit; ++s2)
            ssum += partial[(size_t)s2 * ROWS * C + (size_t)row0 * C + i];
        yt[i] = ssum;
    }
    __syncthreads();

    const int lane  = tid & 31;
    const int wave  = tid >> 5;
    const int ln    = lane & 15;     // N index (and M index for A)
    const int h     = lane >> 4;     // half-wave select
    const int rbase = wave * 16;     // 16 rows per wave

    v8f acc0 = {0.f,0.f,0.f,0.f,0.f,0.f,0.f,0.f};
    v8f acc1 = {0.f,0.f,0.f,0.f,0.f,0.f,0.f,0.f};

#pragma unroll
    for (int kk = 0; kk < C; kk += 4) {
        const int k0 = kk + 2 * h;
        v2f a, b0, b1;
        a.x  = yt[(rbase + ln) * C + k0];
        a.y  = yt[(rbase + ln) * C + k0 + 1];
        b0.x = W[ln * C + k0];              // o = ln
        b0.y = W[ln * C + k0 + 1];
        b1.x = W[(16 + ln) * C + k0];       // o = 16 + ln
        b1.y = W[(16 + ln) * C + k0 + 1];

        acc0 = __builtin_amdgcn_wmma_f32_16x16x4_f32(
            false, a, false, b0, (short)0, acc0, false, false);
        acc1 = __builtin_amdgcn_wmma_f32_16x16x4_f32(
            false, a, false, b1, (short)0, acc1, false, false);
    }

    const float bv0 = bias[ln];
    const float bv1 = bias[16 + ln];

#pragma unroll
    for (int i = 0; i < 8; ++i) {
        const int grow = row0 + rbase + i + 8 * h;   // D: M = i + 8*h
        out[(size_t)grow * O + ln]      = acc0[i] + bv0;
        out[(size_t)grow * O + 16 + ln] = acc1[i] + bv1;
    }
}

// -----------------------------------------------------------------------------
extern "C" void kernel_launch(void* const* d_in, const int* in_sizes, int n_in,
                              void* d_out, int out_size, void* d_ws, size_t ws_size,
                              hipStream_t stream) {
    const float* x     = (const float*)d_in[0];  // (8,1024,1)
    const float* y     = (const float*)d_in[1];  // (8,1024,16)
    const float* t     = (const float*)d_in[2];  // (8,1024,1)
    const float* sigma = (const float*)d_in[3];  // (16)
    const float* W     = (const float*)d_in[4];  // (32,16)
    const float* bias  = (const float*)d_in[5];  // (32)
    float* out         = (float*)d_out;          // (8,1024,32)
    float* partial     = (float*)d_ws;

    // Choose n-split deterministically from ws_size (one slice = 512 KB).
    const size_t per = (size_t)ROWS * C * sizeof(float);
    int nsplit = 8;
    while (nsplit > 1 && (size_t)nsplit * per > ws_size) nsplit >>= 1;
    const int chunk = N / nsplit;

    dim3 g1(M / 256, B, nsplit);
    rbf_partial_kernel<<<g1, 256, 0, stream>>>(x, y, t, sigma, partial, chunk);

    reduce_wmma_kernel<<<ROWS / 128, 256, 0, stream>>>(partial, W, bias, out, nsplit);
}